// M2DM_80161269613410
// MI455X (gfx1250) — compile-verified
//
#include <hip/hip_runtime.h>
#include <math.h>

typedef __attribute__((ext_vector_type(2))) float v2f;
typedef __attribute__((ext_vector_type(8))) float v8f;

#define Bn   32
#define Cn   256
#define HWn  4096        // 64*64
#define AUXn 64
#define RANKn 64
#define HIDn 256
#define OSn  8
#define Dcat 320         // C + AUX

__device__ __forceinline__ float sigmf(float v) { return 1.0f / (1.0f + expf(-v)); }

// ---------------- pass 1: statistics over x ----------------

// g[b,c] = mean_{hw} x[b,c,hw], written into hcat[b*320 + c]
__global__ void k_rowmean(const float* __restrict__ x, float* __restrict__ hcat) {
    int bc = blockIdx.x;                // b*256 + c
    int b = bc >> 8, c = bc & 255;
    const float4* row = (const float4*)(x + (size_t)bc * HWn);
    float s = 0.f;
    for (int t = threadIdx.x; t < HWn / 4; t += 256) {
        float4 v = row[t];
        s += v.x + v.y + v.z + v.w;
    }
    __shared__ float red[256];
    red[threadIdx.x] = s;
    __syncthreads();
    for (int off = 128; off > 0; off >>= 1) {
        if (threadIdx.x < off) red[threadIdx.x] += red[threadIdx.x + off];
        __syncthreads();
    }
    if (threadIdx.x == 0) hcat[b * Dcat + c] = red[0] * (1.0f / (float)HWn);
}

__global__ void k_copy_a(const float* __restrict__ a, float* __restrict__ hcat) {
    int t = blockIdx.x * 256 + threadIdx.x;     // 0 .. 2047
    if (t < Bn * AUXn) {
        int b = t / AUXn, j = t % AUXn;
        hcat[b * Dcat + Cn + j] = a[t];
    }
}

// av[b,hw] = mean_c x[b,c,hw]; mx[b,hw] = max_c x[b,c,hw]
__global__ void k_colstats(const float* __restrict__ x, float* __restrict__ av, float* __restrict__ mx) {
    int b = blockIdx.x >> 4;
    int hw = ((blockIdx.x & 15) << 8) + threadIdx.x;
    const float* base = x + (size_t)b * Cn * HWn + hw;
    float s = 0.f, m = -INFINITY;
    for (int c = 0; c < Cn; ++c) {
        float v = base[(size_t)c * HWn];
        s += v;
        m = fmaxf(m, v);
    }
    av[b * HWn + hw] = s * (1.0f / (float)Cn);
    mx[b * HWn + hw] = m;
}

// ---------------- generic fp32 WMMA GEMM: out = act(A(32xK) @ W(KxN) + bias) ----------------
// One wave per 16-wide column stripe; computes BOTH 16-row tiles (M=32) so each
// B fragment feeds two v_wmma_f32_16x16x4_f32 ops.
// A-layout (16x4 f32): lane l holds row (l&15), VGPR0/1 = K pair at 2*(l>>4).
// B-layout (4x16 f32): lane l holds col (l&15), VGPR0/1 = K pair at 2*(l>>4).
// C/D layout: VGPR v holds row v + 8*(l>>4), col (l&15).
// Out-of-range columns: address is CLAMPED (in-bounds garbage) — column n of D
// lives only in the lane owning n, so garbage never leaks into stored columns.
__global__ void k_gemm_wmma(const float* __restrict__ A, const float* __restrict__ W,
                            const float* __restrict__ bias, float* __restrict__ out,
                            int N, int K, int lda, int ldw, int ldo, int act) {
    int n0 = blockIdx.x * 16;
    int lane = threadIdx.x;            // blockDim.x == 32 (one wave)
    int lm = lane & 15;
    int kg = lane >> 4;                // 0 or 1 -> K sub-pair
    int col = n0 + lm;
    bool colOK = (col < N);
    int colc = colOK ? col : (N - 1);  // clamped, always in-bounds

    float bv = colOK ? bias[col] : 0.0f;
    v8f acc0, acc1;
#pragma unroll
    for (int i = 0; i < 8; ++i) { acc0[i] = bv; acc1[i] = bv; }

    const float* Arow0 = A + (size_t)lm * lda;         // rows 0..15
    const float* Arow1 = A + (size_t)(16 + lm) * lda;  // rows 16..31
    const float* Wcp = W + colc;
    for (int k = 0; k < K; k += 4) {
        int ka = k + (kg << 1);
        v2f a0, a1, vb;
        a0.x = Arow0[ka];
        a0.y = Arow0[ka + 1];
        a1.x = Arow1[ka];
        a1.y = Arow1[ka + 1];
        vb.x = Wcp[(size_t)ka * ldw];
        vb.y = Wcp[(size_t)(ka + 1) * ldw];
        acc0 = __builtin_amdgcn_wmma_f32_16x16x4_f32(false, a0, false, vb,
                                                     (short)0, acc0, false, false);
        acc1 = __builtin_amdgcn_wmma_f32_16x16x4_f32(false, a1, false, vb,
                                                     (short)0, acc1, false, false);
    }

    if (act == 1) {
#pragma unroll
        for (int i = 0; i < 8; ++i) { acc0[i] = fmaxf(acc0[i], 0.0f); acc1[i] = fmaxf(acc1[i], 0.0f); }
    }
    if (colOK) {
#pragma unroll
        for (int v = 0; v < 8; ++v) {
            int row = v + (kg << 3);
            out[(size_t)row * ldo + col] = acc0[v];
            out[(size_t)(row + 16) * ldo + col] = acc1[v];
        }
    }
}

// ---------------- fourier features ----------------
// phi[b, j]    = sin(u[b, j] * w[j]),   j < 32
// phi[b, 32+j] = cos(u[b, j] * w[j])
__global__ void k_fourier(const float* __restrict__ u, float* __restrict__ phi) {
    int b = blockIdx.x, j = threadIdx.x;   // 64 threads
    const float PI = 3.14159265358979323846f;
    int jj = (j < 32) ? j : (j - 32);
    float w = 1.0f + (PI - 1.0f) * ((float)jj / 31.0f);
    float uv = u[b * RANKn + jj] * w;
    phi[b * RANKn + j] = (j < 32) ? sinf(uv) : cosf(uv);
}

// ---------------- gate scalars ----------------
// sigc[b,c] = sigmoid(m_base_c[b,c] * sum_r Wc[b, c*64+r]*phi_c[b,r])
// p[b,s]    = m_base_s[b,s] * sum_r Ws[b, s*64+r]*phi_s[b,r]   (s<8)
__global__ void k_gates(const float* __restrict__ Wc, const float* __restrict__ Ws,
                        const float* __restrict__ phic, const float* __restrict__ phis,
                        const float* __restrict__ mbc, const float* __restrict__ mbs,
                        float* __restrict__ sigc, float* __restrict__ p_arr) {
    int b = blockIdx.x, c = threadIdx.x;   // 256 threads
    __shared__ float pc[RANKn], ps[RANKn];
    if (c < RANKn) pc[c] = phic[b * RANKn + c];
    else if (c < 2 * RANKn) ps[c - RANKn] = phis[b * RANKn + (c - RANKn)];
    __syncthreads();

    const float* wrow = Wc + (size_t)b * (Cn * RANKn) + (size_t)c * RANKn;
    float dot = 0.f;
#pragma unroll 8
    for (int r = 0; r < RANKn; ++r) dot += wrow[r] * pc[r];
    sigc[b * Cn + c] = sigmf(mbc[b * Cn + c] * dot);

    if (c < OSn) {
        const float* wsr = Ws + (size_t)b * (OSn * RANKn) + (size_t)c * RANKn;
        float d2 = 0.f;
#pragma unroll 8
        for (int r = 0; r < RANKn; ++r) d2 += wsr[r] * ps[r];
        p_arr[b * OSn + c] = mbs[b * OSn + c] * d2;
    }
}

// ---------------- spatial gate: 5-point stencil on (av, mx), zero padding ----------------
__global__ void k_conv(const float* __restrict__ av, const float* __restrict__ mx,
                       const float* __restrict__ p_arr, float* __restrict__ sigs) {
    int b = blockIdx.x >> 4;
    int hw = ((blockIdx.x & 15) << 8) + threadIdx.x;
    int i = hw >> 6, j = hw & 63;
    float acc = 0.f;
#pragma unroll
    for (int k2 = 0; k2 < 2; ++k2) {
        const float* y = (k2 == 0 ? av : mx) + b * HWn;
        float p0 = p_arr[b * OSn + k2 * 4 + 0];
        float p1 = p_arr[b * OSn + k2 * 4 + 1];
        float p2 = p_arr[b * OSn + k2 * 4 + 2];
        float p3 = p_arr[b * OSn + k2 * 4 + 3];
        float ctr = -(p0 + p1 + p2 + p3);
        float up = (i > 0)  ? y[hw - 64] : 0.f;   // y[i-1, j]
        float dn = (i < 63) ? y[hw + 64] : 0.f;   // y[i+1, j]
        float lf = (j > 0)  ? y[hw - 1]  : 0.f;   // y[i, j-1]
        float rt = (j < 63) ? y[hw + 1]  : 0.f;   // y[i, j+1]
        acc += p0 * up + p1 * dn + p2 * lf + p3 * rt + ctr * y[hw];
    }
    sigs[b * HWn + hw] = sigmf(acc);
}

// ---------------- final fused scale: out = x * (1 + sigc[b,c] + sigs[b,hw]) ----------------
__global__ void k_final(const float* __restrict__ x, const float* __restrict__ sigc,
                        const float* __restrict__ sigs, float* __restrict__ out) {
    int bc = blockIdx.x;
    int b = bc >> 8;
    float sc = 1.0f + sigc[bc];
    const float4* xr = (const float4*)(x + (size_t)bc * HWn);
    const float4* sr = (const float4*)(sigs + (size_t)b * HWn);
    float4* orow = (float4*)(out + (size_t)bc * HWn);
    for (int t = threadIdx.x; t < HWn / 4; t += 256) {
        float4 xv = xr[t];
        float4 sv = sr[t];
        float4 o;
        o.x = xv.x * (sc + sv.x);
        o.y = xv.y * (sc + sv.y);
        o.z = xv.z * (sc + sv.z);
        o.w = xv.w * (sc + sv.w);
        orow[t] = o;
    }
}

extern "C" void kernel_launch(void* const* d_in, const int* in_sizes, int n_in,
                              void* d_out, int out_size, void* d_ws, size_t ws_size,
                              hipStream_t stream) {
    const float* x     = (const float*)d_in[0];
    const float* a     = (const float*)d_in[1];
    const float* W_pre = (const float*)d_in[2];
    const float* b_pre = (const float*)d_in[3];
    const float* W_bc  = (const float*)d_in[4];
    const float* b_bc  = (const float*)d_in[5];
    const float* W_bs  = (const float*)d_in[6];
    const float* b_bs  = (const float*)d_in[7];
    const float* W_pc  = (const float*)d_in[8];
    const float* b_pc  = (const float*)d_in[9];
    const float* W_ps  = (const float*)d_in[10];
    const float* b_ps  = (const float*)d_in[11];
    const float* W_Wc  = (const float*)d_in[12];
    const float* b_Wc  = (const float*)d_in[13];
    const float* W_Ws  = (const float*)d_in[14];
    const float* b_Ws  = (const float*)d_in[15];
    float* out = (float*)d_out;

    // workspace layout (floats)
    float* ws = (float*)d_ws;
    float* hcat    = ws;                 // 32*320   = 10240
    float* h       = hcat + 10240;       // 32*256   = 8192
    float* av      = h + 8192;           // 32*4096  = 131072
    float* mx      = av + 131072;        // 131072
    float* mbase_c = mx + 131072;        // 8192
    float* mbase_s = mbase_c + 8192;     // 256 (32*8)
    float* u_c     = mbase_s + 256;      // 2048
    float* u_s     = u_c + 2048;         // 2048
    float* phi_c   = u_s + 2048;         // 2048
    float* phi_s   = phi_c + 2048;       // 2048
    float* sig_c   = phi_s + 2048;       // 8192
    float* p_arr   = sig_c + 8192;       // 256
    float* sig_s   = p_arr + 256;        // 131072
    float* Wc_full = sig_s + 131072;     // 32*16384 = 524288
    float* Ws_full = Wc_full + 524288;   // 32*512   = 16384

    // pass 1: statistics (two full reads of x)
    k_rowmean<<<Bn * Cn, 256, 0, stream>>>(x, hcat);
    k_copy_a<<<8, 256, 0, stream>>>(a, hcat);
    k_colstats<<<Bn * 16, 256, 0, stream>>>(x, av, mx);

    // MLP via fp32 WMMA (one wave per 16-wide column stripe, full M=32)
    // h = relu(hcat @ W_pre + b_pre)         K=320 N=256
    k_gemm_wmma<<<16, 32, 0, stream>>>(hcat, W_pre, b_pre, h, 256, 320, Dcat, 256, 256, 1);
    // m_base_c = h @ W_bc + b_bc             N=256
    k_gemm_wmma<<<16, 32, 0, stream>>>(h, W_bc, b_bc, mbase_c, 256, 256, 256, 256, 256, 0);
    // m_base_s = h @ W_bs + b_bs             N=8 (clamped columns)
    k_gemm_wmma<<<1, 32, 0, stream>>>(h, W_bs, b_bs, mbase_s, 8, 256, 256, 8, 8, 0);
    // u_c = h @ W_pc + b_pc                  N=64
    k_gemm_wmma<<<4, 32, 0, stream>>>(h, W_pc, b_pc, u_c, 64, 256, 256, 64, 64, 0);
    // u_s = h @ W_ps + b_ps                  N=64
    k_gemm_wmma<<<4, 32, 0, stream>>>(h, W_ps, b_ps, u_s, 64, 256, 256, 64, 64, 0);
    // Wc = h @ W_Wc + b_Wc                   N=16384
    k_gemm_wmma<<<1024, 32, 0, stream>>>(h, W_Wc, b_Wc, Wc_full, 16384, 256, 256, 16384, 16384, 0);
    // Ws = h @ W_Ws + b_Ws                   N=512
    k_gemm_wmma<<<32, 32, 0, stream>>>(h, W_Ws, b_Ws, Ws_full, 512, 256, 256, 512, 512, 0);

    // fourier features
    k_fourier<<<Bn, 64, 0, stream>>>(u_c, phi_c);
    k_fourier<<<Bn, 64, 0, stream>>>(u_s, phi_s);

    // per-channel + stencil gates
    k_gates<<<Bn, 256, 0, stream>>>(Wc_full, Ws_full, phi_c, phi_s, mbase_c, mbase_s, sig_c, p_arr);
    k_conv<<<Bn * 16, 256, 0, stream>>>(av, mx, p_arr, sig_s);

    // final fused output (third read of x should hit 192MB L2)
    k_final<<<Bn * Cn, 256, 0, stream>>>(x, sig_c, sig_s, out);
}